// GraphConvolutionLayer_51737176047901
// MI455X (gfx1250) — compile-verified
//
#include <hip/hip_runtime.h>
#include <hip/hip_bf16.h>

// ---------------------------------------------------------------------------
// GraphConvolutionLayer for MI455X (gfx1250, wave32)
//   out = segment_sum( (x @ W)[src] * edge_attr, dst ) + bias
//
// Phase 1: bias-seed the output (folds "+bias" into the scatter identity).
// Phase 2: WMMA f32 16x16x4 GEMM  x[100000,128] @ W[128,64] -> ws[100000,64]
// Phase 3: wave-per-edge gather-multiply-scatter with f32 HW atomics.
// ---------------------------------------------------------------------------

typedef __attribute__((ext_vector_type(2))) float v2f;
typedef __attribute__((ext_vector_type(8))) float v8f;

#define IN_F  128
#define OUT_F 64

// ----------------------------- Phase 1: seed -------------------------------
__global__ void gc_seed_bias(float* __restrict__ out,
                             const float* __restrict__ bias,
                             int total /* = n_nodes*OUT_F */) {
    int i = blockIdx.x * blockDim.x + threadIdx.x;
    if (i < total) {
        out[i] = bias[i & (OUT_F - 1)];
    }
}

// ----------------------------- Phase 2: GEMM -------------------------------
// One wave32 computes a 16x64 tile of xw. K=128 -> 32 steps of 16x16x4.
// A-matrix f32 16x4 layout: lane L holds row m=L&15, K-pair kb=(L>>4)*2 in
//   VGPRs {a.x = K=kb, a.y = K=kb+1}.
// B-matrix f32 4x16 layout (mirror): lane L holds col n=L&15, rows kb,kb+1.
// C/D f32 16x16 layout: VGPR v -> row v + (L>>4)*8, col L&15.
__global__ void __launch_bounds__(256)
gc_gemm_wmma(const float* __restrict__ x,
             const float* __restrict__ w,
             float* __restrict__ xw,
             int n_nodes) {
    const int wave  = (blockIdx.x * blockDim.x + threadIdx.x) >> 5;
    const int lane  = threadIdx.x & 31;
    const int tileM = wave * 16;
    if (tileM >= n_nodes) return;           // whole wave exits together -> EXEC stays all-1s for WMMA

    const int m  = lane & 15;               // row within tile (A) / col within tile (B,C,D)
    const int kb = (lane >> 4) << 1;        // 0 or 2: which K-pair this half-wave holds

    const float* __restrict__ xrow = x + (size_t)(tileM + m) * IN_F;

    v8f acc0 = {}, acc1 = {}, acc2 = {}, acc3 = {};

    #pragma unroll 4
    for (int kk = 0; kk < IN_F; kk += 4) {
        v2f a;
        a.x = xrow[kk + kb];
        a.y = xrow[kk + kb + 1];

        const float* __restrict__ w0 = w + (size_t)(kk + kb) * OUT_F + m;
        const float* __restrict__ w1 = w0 + OUT_F;

        v2f b0; b0.x = w0[0];  b0.y = w1[0];
        v2f b1; b1.x = w0[16]; b1.y = w1[16];
        v2f b2; b2.x = w0[32]; b2.y = w1[32];
        v2f b3; b3.x = w0[48]; b3.y = w1[48];

        acc0 = __builtin_amdgcn_wmma_f32_16x16x4_f32(false, a, false, b0, (short)0, acc0, false, false);
        acc1 = __builtin_amdgcn_wmma_f32_16x16x4_f32(false, a, false, b1, (short)0, acc1, false, false);
        acc2 = __builtin_amdgcn_wmma_f32_16x16x4_f32(false, a, false, b2, (short)0, acc2, false, false);
        acc3 = __builtin_amdgcn_wmma_f32_16x16x4_f32(false, a, false, b3, (short)0, acc3, false, false);
    }

    // Store D: VGPR v -> row (tileM + v + (lane>>4)*8), col (lane&15) + 16*nt
    const int rowBase = tileM + ((lane >> 4) << 3);
    float* __restrict__ obase = xw + (size_t)rowBase * OUT_F + m;
    #pragma unroll
    for (int v = 0; v < 8; ++v) {
        float* __restrict__ orow = obase + (size_t)v * OUT_F;
        orow[0]  = acc0[v];
        orow[16] = acc1[v];
        orow[32] = acc2[v];
        orow[48] = acc3[v];
    }
}

// ------------------------- Phase 3: edge scatter ---------------------------
// One wave32 per edge; lane c owns features [2c, 2c+1] (float2).
// Gather xw[src] (L2-resident, 256B contiguous per edge), multiply by
// edge_attr[e] (streamed), scatter-add into out[dst] with f32 HW atomics.
__global__ void __launch_bounds__(256)
gc_edge_scatter(const float* __restrict__ xw,
                const int*   __restrict__ edge_index,  // [2, E]: src row, dst row
                const float* __restrict__ edge_attr,   // [E, OUT_F]
                float*       __restrict__ out,
                int n_edges) {
    const long long tid = (long long)blockIdx.x * blockDim.x + threadIdx.x;
    const int e    = (int)(tid >> 5);
    const int lane = (int)(tid & 31);
    if (e >= n_edges) return;

    const int src = edge_index[e];
    const int dst = edge_index[n_edges + e];

    const float2* __restrict__ g = (const float2*)(xw        + (size_t)src * OUT_F);
    const float2* __restrict__ a = (const float2*)(edge_attr + (size_t)e   * OUT_F);

    const float2 xv = g[lane];
    const float2 av = a[lane];

    float* __restrict__ o = out + (size_t)dst * OUT_F + (lane << 1);
    unsafeAtomicAdd(o + 0, xv.x * av.x);   // -> global_atomic_add_f32
    unsafeAtomicAdd(o + 1, xv.y * av.y);
}

// ------------------------------- Launcher ----------------------------------
extern "C" void kernel_launch(void* const* d_in, const int* in_sizes, int n_in,
                              void* d_out, int out_size, void* d_ws, size_t ws_size,
                              hipStream_t stream) {
    const float* x          = (const float*)d_in[0];   // [N, 128]
    const int*   edge_index = (const int*)  d_in[1];   // [2, E]
    const float* edge_attr  = (const float*)d_in[2];   // [E, 64]
    const float* weight     = (const float*)d_in[3];   // [128, 64]
    const float* bias       = (const float*)d_in[4];   // [64]
    float*       out        = (float*)d_out;           // [N, 64]

    const int n_nodes = in_sizes[0] / IN_F;
    const int n_edges = in_sizes[1] / 2;

    float* xw = (float*)d_ws;                          // [N, 64] scratch (25.6 MB)

    // Phase 1: out[n,c] = bias[c]
    {
        const int total = n_nodes * OUT_F;
        const int blk = 256;
        const int grd = (total + blk - 1) / blk;
        gc_seed_bias<<<grd, blk, 0, stream>>>(out, bias, total);
    }

    // Phase 2: xw = x @ W via WMMA (wave per 16-row tile, 8 waves/block)
    {
        const int tiles = (n_nodes + 15) / 16;                 // 6250
        const int wavesPerBlock = 8;                           // 256 threads
        const int grd = (tiles + wavesPerBlock - 1) / wavesPerBlock;
        gc_gemm_wmma<<<grd, 256, 0, stream>>>(x, weight, xw, n_nodes);
    }

    // Phase 3: wave per edge
    {
        const long long threads = (long long)n_edges * 32;
        const int blk = 256;
        const long long grd = (threads + blk - 1) / blk;       // 125000 blocks
        gc_edge_scatter<<<(int)grd, blk, 0, stream>>>(xw, edge_index, edge_attr, out, n_edges);
    }
}